// Model_68384469287342
// MI455X (gfx1250) — compile-verified
//
#include <hip/hip_runtime.h>
#include <hip/hip_bf16.h>
#include <math.h>

// Model dims
#define BATCH 32
#define LSEQ  512
#define NVAR  256
#define NFEAT 7
#define TOK   263            // NVAR + NFEAT tokens per batch
#define DMODEL 256
#define DSTATE 16
#define DTRANK 16
#define DINNER 256
#define DFF    512
#define PLEN   96
#define BT     (BATCH * TOK) // 8416 tokens (multiple of 16)

typedef __attribute__((ext_vector_type(2))) float v2f;
typedef __attribute__((ext_vector_type(8))) float v8f;
typedef __attribute__((ext_vector_type(4))) unsigned int v4u;
typedef __attribute__((ext_vector_type(8))) int v8i;
typedef __attribute__((ext_vector_type(4))) int v4i;

#ifndef __has_builtin
#define __has_builtin(x) 0
#endif

// ---------------------------------------------------------------------------
// WMMA fp32 GEMM:  C[M,N] = epi(A[M,K] @ W + bias)
// W element (k,n) = Bw[k*bsk + n*bsn].  BK1 => bsk==1 (K-contiguous storage).
// Block = 256 threads = 8 waves; each wave owns a 16 x (NT*16) strip.
// NT is compile-time => branch-free K loop, EXEC all-ones at every WMMA.
// K is a multiple of 8.  EPI: 0 = none, 1 = relu, 2 = softplus.
// A TDM tensor_load_to_lds stages the block's first 16x64 weight lines into
// LDS as a prefetch (compute still reads the global path).
// ---------------------------------------------------------------------------
template <int NT, int EPI, bool BK1>
__global__ __launch_bounds__(256) void gemm_wmma_f32(
    const float* __restrict__ A, int lda,
    const float* __restrict__ Bw, int bsk, int bsn,
    const float* __restrict__ bias,
    float* __restrict__ C, int ldc,
    int M, int K)
{
    __shared__ float tdm_stage[16 * 64];    // TDM staging / prefetch target

    const int lane = threadIdx.x & 31;
    const int wave = threadIdx.x >> 5;
    const int m0 = (blockIdx.y * 8 + wave) * 16;
    const int n0 = blockIdx.x * (NT * 16);
    const int row = lane & 15;   // M row (A), N col (B/C/D)
    const int kh  = lane >> 4;   // selects K pair {0,1} vs {2,3}

#if __has_builtin(__builtin_amdgcn_tensor_load_to_lds)
    if (wave == 0) {
        // D# for a 2D tile: 16 lines x min(64,line) elems of this block's panel
        const unsigned long long gaddr =
            (unsigned long long)(const void*)(Bw + (size_t)n0 * (BK1 ? bsn : 1));
        const unsigned lds_lo = (unsigned)(unsigned long long)(__SIZE_TYPE__)&tdm_stage[0];
        const unsigned lstride = (unsigned)(BK1 ? bsn : bsk);  // elems between lines
        const unsigned tile0   = lstride < 64u ? lstride : 64u;
        v4u g0;
        g0.x = 1u;                                    // count=1, user descriptor
        g0.y = lds_lo;                                // lds_addr
        g0.z = (unsigned)gaddr;                       // global_addr[31:0]
        g0.w = ((unsigned)(gaddr >> 32) & 0x01FFFFFFu) | (2u << 30);  // addr[56:32], type=2
        v8i g1;
        g1[0] = (int)(2u << 16);                      // data_size = 4 bytes
        g1[1] = (int)((lstride & 0xFFFFu) << 16);     // tensor_dim0[15:0]
        g1[2] = (int)((lstride >> 16) | (16u << 16)); // tensor_dim0[31:16] | tensor_dim1[15:0]=16
        g1[3] = (int)(tile0 << 16);                   // tensor_dim1[31:16]=0 | tile_dim0
        g1[4] = 16;                                   // tile_dim1 = 16 lines
        g1[5] = (int)lstride;                         // tensor_dim0_stride[31:0]
        g1[6] = 0;                                    // stride0[47:32] | stride1[15:0]
        g1[7] = 0;
        const v4i z4 = (v4i)0;
#if defined(__clang_major__) && __clang_major__ >= 23
        const v8i z8 = (v8i)0;
        __builtin_amdgcn_tensor_load_to_lds(g0, g1, z4, z4, z8, 0);
#else
        __builtin_amdgcn_tensor_load_to_lds(g0, g1, z4, z4, 0);
#endif
    }
#endif

    if (m0 < M) {
        v8f acc[NT] = {};

        const float* Ar = A + (size_t)(m0 + row) * lda + 2 * kh;
        const float* Bj[NT];
#pragma unroll
        for (int j = 0; j < NT; ++j)
            Bj[j] = Bw + (size_t)(n0 + j * 16 + row) * bsn + (size_t)(2 * kh) * bsk;

        for (int k0 = 0; k0 < K; k0 += 8) {
            // issue all loads of this body first, then 2*NT WMMAs
            const v2f a0 = *(const v2f*)(Ar + k0);
            const v2f a1 = *(const v2f*)(Ar + k0 + 4);
            v2f b0[NT], b1[NT];
#pragma unroll
            for (int j = 0; j < NT; ++j) {
                if (BK1) {
                    b0[j] = *(const v2f*)(Bj[j] + k0);
                    b1[j] = *(const v2f*)(Bj[j] + k0 + 4);
                } else {
                    const float* bp = Bj[j] + (size_t)k0 * bsk;
                    b0[j].x = bp[0];
                    b0[j].y = bp[bsk];
                    b1[j].x = bp[4 * (size_t)bsk];
                    b1[j].y = bp[5 * (size_t)bsk];
                }
            }
#pragma unroll
            for (int j = 0; j < NT; ++j)
                acc[j] = __builtin_amdgcn_wmma_f32_16x16x4_f32(
                    false, a0, false, b0[j], (short)0, acc[j], false, false);
#pragma unroll
            for (int j = 0; j < NT; ++j)
                acc[j] = __builtin_amdgcn_wmma_f32_16x16x4_f32(
                    false, a1, false, b1[j], (short)0, acc[j], false, false);
        }

#pragma unroll
        for (int j = 0; j < NT; ++j) {
            const int n = n0 + j * 16 + row;
            const float bv = bias ? bias[n] : 0.0f;
#pragma unroll
            for (int v = 0; v < 8; ++v) {
                const int m = m0 + v + 8 * kh;
                float val = acc[j][v] + bv;
                if (EPI == 1) val = fmaxf(val, 0.0f);
                if (EPI == 2) val = (val > 20.0f) ? val : log1pf(__expf(val));
                C[(size_t)m * ldc + n] = val;
            }
        }
    }

#if __has_builtin(__builtin_amdgcn_s_wait_tensorcnt)
    if (wave == 0) __builtin_amdgcn_s_wait_tensorcnt(0);
#endif
}

// ---------------------------------------------------------------------------
// Build inverted embedding input  emb_in[b, tok, l]
// ---------------------------------------------------------------------------
__global__ void build_emb_in_k(const float* __restrict__ x_enc,
                               const float* __restrict__ x_mark,
                               float* __restrict__ out)
{
    size_t idx = (size_t)blockIdx.x * blockDim.x + threadIdx.x;
    const size_t total = (size_t)BT * LSEQ;
    if (idx >= total) return;
    const int l   = (int)(idx % LSEQ);
    const int tkn = (int)((idx / LSEQ) % TOK);
    const int b   = (int)(idx / ((size_t)LSEQ * TOK));
    float v;
    if (tkn < NVAR) {
        v = x_enc[((size_t)b * LSEQ + l) * NVAR + tkn];
        if (v == -9999.0f) v = -1.0f;
    } else {
        v = x_mark[((size_t)b * LSEQ + l) * NFEAT + (tkn - NVAR)];
    }
    out[idx] = v;
}

// ---------------------------------------------------------------------------
// Causal depthwise conv (kernel 2) + SiLU.  xr = xz[..., 0:DINNER]
// ---------------------------------------------------------------------------
__global__ void conv_silu_k(const float* __restrict__ xz,
                            const float* __restrict__ cw,
                            const float* __restrict__ cb,
                            float* __restrict__ xc)
{
    size_t idx = (size_t)blockIdx.x * blockDim.x + threadIdx.x;
    const size_t total = (size_t)BT * DINNER;
    if (idx >= total) return;
    const int d  = (int)(idx % DINNER);
    const size_t bt = idx / DINNER;
    const int t  = (int)(bt % TOK);
    const float cur  = xz[bt * (2 * DINNER) + d];
    const float prev = (t > 0) ? xz[(bt - 1) * (2 * DINNER) + d] : 0.0f;
    const float v = cb[d] + prev * cw[d * 2 + 0] + cur * cw[d * 2 + 1];
    xc[idx] = v / (1.0f + __expf(-v));   // silu
}

// ---------------------------------------------------------------------------
// Selective scan + D skip + SiLU(z) gating.  Block per batch, thread per chan.
// ---------------------------------------------------------------------------
__global__ __launch_bounds__(256) void scan_gate_k(
    const float* __restrict__ xc,
    const float* __restrict__ dt,
    const float* __restrict__ xdbl,   // [BT,48]: [dt_r(16) | B(16) | C(16)]
    const float* __restrict__ xz,     // z = xz[..., DINNER:2*DINNER]
    const float* __restrict__ A_log,  // [DINNER, DSTATE]
    const float* __restrict__ Dp,     // [DINNER]
    float* __restrict__ y)
{
    const int b = blockIdx.x;
    const int d = threadIdx.x;

    float Arow[DSTATE], hst[DSTATE];
#pragma unroll
    for (int s = 0; s < DSTATE; ++s) {
        Arow[s] = -__expf(A_log[d * DSTATE + s]);
        hst[s]  = 0.0f;
    }
    const float Dd = Dp[d];

    __shared__ float sB[DSTATE];
    __shared__ float sC[DSTATE];

    for (int t = 0; t < TOK; ++t) {
        const size_t bt = (size_t)b * TOK + t;
        __syncthreads();                       // protect sB/sC from prev step
        if (threadIdx.x < 32) {
            const float v = xdbl[bt * 48 + 16 + threadIdx.x];
            if (threadIdx.x < 16) sB[threadIdx.x] = v;
            else                  sC[threadIdx.x - 16] = v;
        }
        __syncthreads();
        // prefetch next step's B/C vector (global_prefetch_b8)
        if (t + 1 < TOK) __builtin_prefetch(&xdbl[(bt + 1) * 48 + 16], 0, 1);

        const float u   = xc[bt * DINNER + d];
        const float dtv = dt[bt * DINNER + d];
        const float du  = dtv * u;
        float acc = 0.0f;
#pragma unroll
        for (int s = 0; s < DSTATE; ++s) {
            const float dA = __expf(dtv * Arow[s]);
            hst[s] = dA * hst[s] + du * sB[s];
            acc += hst[s] * sC[s];
        }
        const float z = xz[bt * (2 * DINNER) + DINNER + d];
        const float gate = z / (1.0f + __expf(-z));
        y[bt * DINNER + d] = (acc + u * Dd) * gate;
    }
}

// ---------------------------------------------------------------------------
// LayerNorm over DMODEL with optional residual:  out = LN(x + r) * g + b
// ---------------------------------------------------------------------------
__global__ __launch_bounds__(256) void ln_residual_k(
    const float* __restrict__ x, const float* __restrict__ r,
    const float* __restrict__ g, const float* __restrict__ bb,
    float* __restrict__ out)
{
    const size_t t = blockIdx.x;
    const int d = threadIdx.x;
    float v = x[t * DMODEL + d];
    if (r) v += r[t * DMODEL + d];
    __shared__ float s1[DMODEL];
    __shared__ float s2[DMODEL];
    s1[d] = v; s2[d] = v * v;
    __syncthreads();
    for (int off = DMODEL / 2; off > 0; off >>= 1) {
        if (d < off) { s1[d] += s1[d + off]; s2[d] += s2[d + off]; }
        __syncthreads();
    }
    const float mean = s1[0] * (1.0f / DMODEL);
    const float var  = s2[0] * (1.0f / DMODEL) - mean * mean;
    const float inv  = rsqrtf(var + 1e-5f);
    out[t * DMODEL + d] = (v - mean) * inv * g[d] + bb[d];
}

__global__ void reverse_copy_k(const float* __restrict__ in, float* __restrict__ out)
{
    size_t idx = (size_t)blockIdx.x * blockDim.x + threadIdx.x;
    const size_t total = (size_t)BT * DMODEL;
    if (idx >= total) return;
    const int d = (int)(idx % DMODEL);
    const size_t bt = idx / DMODEL;
    const int t = (int)(bt % TOK);
    const int b = (int)(bt / TOK);
    out[idx] = in[((size_t)b * TOK + (TOK - 1 - t)) * DMODEL + d];
}

__global__ void reverse_add_k(float* __restrict__ acc, const float* __restrict__ in)
{
    size_t idx = (size_t)blockIdx.x * blockDim.x + threadIdx.x;
    const size_t total = (size_t)BT * DMODEL;
    if (idx >= total) return;
    const int d = (int)(idx % DMODEL);
    const size_t bt = idx / DMODEL;
    const int t = (int)(bt % TOK);
    const int b = (int)(bt / TOK);
    acc[idx] += in[((size_t)b * TOK + (TOK - 1 - t)) * DMODEL + d];
}

// p [B, TOK, PLEN] -> out [B, PLEN, NVAR]:  out[b,pl,n] = p[b,n,pl]
__global__ void write_out_k(const float* __restrict__ p, float* __restrict__ out)
{
    size_t idx = (size_t)blockIdx.x * blockDim.x + threadIdx.x;
    const size_t total = (size_t)BATCH * PLEN * NVAR;
    if (idx >= total) return;
    const int n  = (int)(idx % NVAR);
    const int pl = (int)((idx / NVAR) % PLEN);
    const int b  = (int)(idx / ((size_t)NVAR * PLEN));
    out[idx] = p[((size_t)b * TOK + n) * PLEN + pl];
}

// ---------------------------------------------------------------------------
// Host-side helpers
// ---------------------------------------------------------------------------
static void launch_gemm(const float* A, int lda, const float* Bw, int bsk, int bsn,
                        const float* bias, float* C, int ldc,
                        int M, int N, int K, int epi, hipStream_t s)
{
    const bool bk1 = (bsk == 1);
    const int nt = (N % 64 == 0) ? 4 : 3;      // every launch: N % (nt*16) == 0
    dim3 grid(N / (nt * 16), (((M + 15) / 16) + 7) / 8);
    dim3 blk(256);
    if (nt == 4) {
        if (bk1) {
            if (epi == 1)
                gemm_wmma_f32<4, 1, true><<<grid, blk, 0, s>>>(A, lda, Bw, bsk, bsn, bias, C, ldc, M, K);
            else if (epi == 2)
                gemm_wmma_f32<4, 2, true><<<grid, blk, 0, s>>>(A, lda, Bw, bsk, bsn, bias, C, ldc, M, K);
            else
                gemm_wmma_f32<4, 0, true><<<grid, blk, 0, s>>>(A, lda, Bw, bsk, bsn, bias, C, ldc, M, K);
        } else {
            gemm_wmma_f32<4, 0, false><<<grid, blk, 0, s>>>(A, lda, Bw, bsk, bsn, bias, C, ldc, M, K);
        }
    } else {
        gemm_wmma_f32<3, 0, true><<<grid, blk, 0, s>>>(A, lda, Bw, bsk, bsn, bias, C, ldc, M, K);
    }
}

// Param flattening (dict insertion order):
// 0 x_enc, 1 x_mark_enc, 2 x_dec, 3 x_mark_dec, 4 emb_w, 5 emb_b,
// layer l base = 6 + 26*l:
//   +0..8  : m1 {in_proj, conv_w, conv_b, x_proj, dt_w, dt_b, A_log, D, out_proj}
//   +9..17 : m2 (same 9)
//   +18 n1_g, +19 n1_b, +20 n2_g, +21 n2_b, +22 ff1_w, +23 ff1_b, +24 ff2_w, +25 ff2_b
// 58 norm_g, 59 norm_b, 60 proj_w, 61 proj_b
static void run_mamba(const float* x, void* const* d_in, int pb,
                      float* xz, float* xc, float* xdbl, float* dtb,
                      float* yb, float* mout, hipStream_t s)
{
    const float* in_proj  = (const float*)d_in[pb + 0];
    const float* conv_w   = (const float*)d_in[pb + 1];
    const float* conv_b   = (const float*)d_in[pb + 2];
    const float* x_proj   = (const float*)d_in[pb + 3];
    const float* dt_w     = (const float*)d_in[pb + 4];
    const float* dt_b     = (const float*)d_in[pb + 5];
    const float* A_log    = (const float*)d_in[pb + 6];
    const float* Dp       = (const float*)d_in[pb + 7];
    const float* out_proj = (const float*)d_in[pb + 8];

    const int eb = 256, ne = (BT * DINNER + eb - 1) / eb;

    // xz = x @ in_proj           [BT, 512], W row-major (bsk=N)
    launch_gemm(x, DMODEL, in_proj, 2 * DINNER, 1, nullptr, xz, 2 * DINNER,
                BT, 2 * DINNER, DMODEL, 0, s);
    conv_silu_k<<<ne, eb, 0, s>>>(xz, conv_w, conv_b, xc);
    // x_dbl = xc @ x_proj.T      [BT, 48], W^T stored -> bsk=1
    launch_gemm(xc, DINNER, x_proj, 1, DINNER, nullptr, xdbl, 48,
                BT, 48, DINNER, 0, s);
    // dt = softplus(x_dbl[:, :16] @ dt_w.T + dt_b)   [BT, 256]
    launch_gemm(xdbl, 48, dt_w, 1, DTRANK, dt_b, dtb, DINNER,
                BT, DINNER, DTRANK, 2, s);
    scan_gate_k<<<BATCH, DINNER, 0, s>>>(xc, dtb, xdbl, xz, A_log, Dp, yb);
    // mout = y @ out_proj        [BT, 256], W row-major
    launch_gemm(yb, DINNER, out_proj, DMODEL, 1, nullptr, mout, DMODEL,
                BT, DMODEL, DINNER, 0, s);
}

extern "C" void kernel_launch(void* const* d_in, const int* in_sizes, int n_in,
                              void* d_out, int out_size, void* d_ws, size_t ws_size,
                              hipStream_t stream)
{
    (void)in_sizes; (void)n_in; (void)out_size; (void)ws_size;

    const float* x_enc  = (const float*)d_in[0];
    const float* x_mark = (const float*)d_in[1];
    const float* emb_w  = (const float*)d_in[4];
    const float* emb_b  = (const float*)d_in[5];

    float* ws = (float*)d_ws;
    size_t o = 0;
    float* R0   = ws + o; o += (size_t)BT * 512;    // emb_in / xz / ffn-mid
    float* h    = ws + o; o += (size_t)BT * DMODEL;
    float* accb = ws + o; o += (size_t)BT * DMODEL;
    float* hrev = ws + o; o += (size_t)BT * DMODEL;
    float* xc   = ws + o; o += (size_t)BT * DINNER;
    float* dtb  = ws + o; o += (size_t)BT * DINNER;
    float* xdbl = ws + o; o += (size_t)BT * 48;
    float* yb   = ws + o; o += (size_t)BT * DINNER; // also reused for proj out
    float* mout = ws + o; o += (size_t)BT * DMODEL;

    const int eb = 256;
    const int nE  = (int)(((size_t)BT * LSEQ + eb - 1) / eb);
    const int nDM = (int)(((size_t)BT * DMODEL + eb - 1) / eb);

    // Inverted embedding: h = emb_in @ emb_w.T + emb_b
    build_emb_in_k<<<nE, eb, 0, stream>>>(x_enc, x_mark, R0);
    launch_gemm(R0, LSEQ, emb_w, 1, LSEQ, emb_b, h, DMODEL, BT, DMODEL, LSEQ, 0, stream);

    for (int l = 0; l < 2; ++l) {
        const int lb = 6 + 26 * l;
        // m1 (forward) -> accb
        run_mamba(h, d_in, lb, R0, xc, xdbl, dtb, yb, accb, stream);
        // m2 (reverse): run on reversed tokens, add reversed output
        reverse_copy_k<<<nDM, eb, 0, stream>>>(h, hrev);
        run_mamba(hrev, d_in, lb + 9, R0, xc, xdbl, dtb, yb, mout, stream);
        reverse_add_k<<<nDM, eb, 0, stream>>>(accb, mout);
        // h = LN(h + acc)
        ln_residual_k<<<BT, DMODEL, 0, stream>>>(h, accb,
            (const float*)d_in[lb + 18], (const float*)d_in[lb + 19], h);
        // FFN: mid = relu(h @ ff1.T + b1); y2 = mid @ ff2.T + b2
        launch_gemm(h, DMODEL, (const float*)d_in[lb + 22], 1, DMODEL,
                    (const float*)d_in[lb + 23], R0, DFF, BT, DFF, DMODEL, 1, stream);
        launch_gemm(R0, DFF, (const float*)d_in[lb + 24], 1, DFF,
                    (const float*)d_in[lb + 25], mout, DMODEL, BT, DMODEL, DFF, 0, stream);
        // h = LN(h + y2)
        ln_residual_k<<<BT, DMODEL, 0, stream>>>(h, mout,
            (const float*)d_in[lb + 20], (const float*)d_in[lb + 21], h);
    }

    // final norm + projection + transpose to [B, PLEN, NVAR]
    ln_residual_k<<<BT, DMODEL, 0, stream>>>(h, nullptr,
        (const float*)d_in[58], (const float*)d_in[59], h);
    launch_gemm(h, DMODEL, (const float*)d_in[60], 1, DMODEL,
                (const float*)d_in[61], yb, PLEN, BT, PLEN, DMODEL, 0, stream);
    const int nO = (int)(((size_t)BATCH * PLEN * NVAR + eb - 1) / eb);
    write_out_k<<<nO, eb, 0, stream>>>(yb, (float*)d_out);
}